// Mamba2_45724221833857
// MI455X (gfx1250) — compile-verified
//
#include <hip/hip_runtime.h>
#include <hip/hip_bf16.h>

typedef __attribute__((ext_vector_type(2))) float v2f;
typedef __attribute__((ext_vector_type(8))) float v8f;
typedef __attribute__((ext_vector_type(4))) unsigned int u32x4;
typedef __attribute__((ext_vector_type(4))) int i32x4;
typedef __attribute__((ext_vector_type(8))) int i32x8;

#define D_MODEL 1024
#define D_INNER 2048
#define D_STATE 64
#define NHEADS  32
#define HEADDIM 64
#define CHUNK   64
#define CONV_DIM 2176            // D_INNER + 2*64
#define D_IN_PROJ 4256           // 2*2048 + 128 + 32
#define BATCH 4
#define SEQ   4096
#define NCHUNK 64                // SEQ / CHUNK
#define NTOK  (BATCH*SEQ)        // 16384

static __device__ __forceinline__ v8f wmma4(v2f a, v2f b, v8f c) {
    // D = A(16x4 f32) * B(4x16 f32) + C(16x16 f32)
    return __builtin_amdgcn_wmma_f32_16x16x4_f32(false, a, false, b, (short)0, c,
                                                 false, false);
}
static __device__ __forceinline__ v8f zero8() {
    v8f z = {0.f,0.f,0.f,0.f,0.f,0.f,0.f,0.f};
    return z;
}
static __device__ __forceinline__ float siluf(float x) {
    return x / (1.f + __expf(-x));
}

// ---------------------------------------------------------------------------
// Tensor Data Mover: DMA a 64x64 fp32 tile (row stride = row_stride_elems)
// from global memory into LDS.  D# layout per CDNA5 ISA §8 (async_tensor).
// ---------------------------------------------------------------------------
#if __has_builtin(__builtin_amdgcn_tensor_load_to_lds) && \
    __has_builtin(__builtin_amdgcn_s_wait_tensorcnt)
#define MAMBA2_HAS_TDM 1
static __device__ __forceinline__ void
tdm_load_tile64(const float* gsrc, unsigned lds_byte_off,
                unsigned row_stride_elems) {
    const unsigned long long ga = (unsigned long long)gsrc;
    u32x4 g0;
    g0[0] = 1u;                                   // count=1, user descriptor
    g0[1] = lds_byte_off;                         // lds_addr
    g0[2] = (unsigned)(ga & 0xffffffffu);         // global_addr[31:0]
    g0[3] = (unsigned)((ga >> 32) & 0x1ffffffu)   // global_addr[56:32]
            | (2u << 30);                         // type = 2 ("image")
    i32x8 g1;
    g1[0] = (2 << 16);          // wg_mask=0, data_size=2 (4 bytes), no pad/iter
    g1[1] = (64 << 16);         // tensor_dim0[15:0] = 64
    g1[2] = (64 << 16);         // tensor_dim0[31:16]=0 | tensor_dim1[15:0]=64
    g1[3] = (64 << 16);         // tensor_dim1[31:16]=0 | tile_dim0 = 64
    g1[4] = 64;                 // tile_dim1 = 64, tile_dim2 = 0
    g1[5] = (int)row_stride_elems;   // tensor_dim0_stride[31:0] (elements)
    g1[6] = 0;                  // stride hi | tensor_dim1_stride lo (unused, 2D)
    g1[7] = 0;
    const i32x4 gz = {0, 0, 0, 0};
#if defined(__clang_major__) && (__clang_major__ >= 23)
    const i32x8 gz8 = {0, 0, 0, 0, 0, 0, 0, 0};
    __builtin_amdgcn_tensor_load_to_lds(g0, g1, gz, gz, gz8, 0);
#else
    __builtin_amdgcn_tensor_load_to_lds(g0, g1, gz, gz, 0);
#endif
}
static __device__ __forceinline__ unsigned lds_off(const void* p) {
    // generic pointer to LDS: low 32 bits are the LDS byte address
    return (unsigned)(unsigned long long)p;
}
#endif

// ---------------------------------------------------------------------------
// fp32 WMMA GEMM: C[M,N] = A[M,K] * W[N,K]^T.  Block = 64 thr (2 waves);
// wave covers 16(M) x 16*TPW(N), no edge guards (launch sizes divide exactly).
// Register double-buffered K loop + global_prefetch on the A row.
// ---------------------------------------------------------------------------
template <int TPW>
__global__ void __launch_bounds__(64)
mamba2_gemm_f32_wmma(const float* __restrict__ A, const float* __restrict__ W,
                     float* __restrict__ C, int M, int N, int K) {
    (void)M;
    const int lane = threadIdx.x & 31;
    const int wave = threadIdx.x >> 5;
    const int half = lane >> 4;
    const int lm   = lane & 15;
    const int row0  = blockIdx.x * 16;
    const int tile0 = (blockIdx.y * 2 + wave) * TPW;

    const float* arow = A + (size_t)(row0 + lm) * K + 2 * half;
    const float* wrow[TPW];
#pragma unroll
    for (int t = 0; t < TPW; ++t)
        wrow[t] = W + (size_t)((tile0 + t) * 16 + lm) * K + 2 * half;

    v8f acc[TPW];
#pragma unroll
    for (int t = 0; t < TPW; ++t) acc[t] = zero8();

    v2f a_c = *(const v2f*)(arow);
    v2f b_c[TPW];
#pragma unroll
    for (int t = 0; t < TPW; ++t) b_c[t] = *(const v2f*)(wrow[t]);

    for (int k0 = 0; k0 + 4 < K; k0 += 4) {
        __builtin_prefetch(arow + k0 + 132, 0, 0);   // global_prefetch_b8
        v2f a_n = *(const v2f*)(arow + k0 + 4);
        v2f b_n[TPW];
#pragma unroll
        for (int t = 0; t < TPW; ++t) b_n[t] = *(const v2f*)(wrow[t] + k0 + 4);
#pragma unroll
        for (int t = 0; t < TPW; ++t) acc[t] = wmma4(a_c, b_c[t], acc[t]);
        a_c = a_n;
#pragma unroll
        for (int t = 0; t < TPW; ++t) b_c[t] = b_n[t];
    }
#pragma unroll
    for (int t = 0; t < TPW; ++t) acc[t] = wmma4(a_c, b_c[t], acc[t]);

#pragma unroll
    for (int t = 0; t < TPW; ++t) {
        const int col = (tile0 + t) * 16 + lm;
#pragma unroll
        for (int r = 0; r < 8; ++r) {
            const int row = row0 + r + 8 * half;
            C[(size_t)row * N + col] = acc[t][r];
        }
    }
}

// ---------------------------------------------------------------------------
// Causal depthwise conv (width 4) + bias + SiLU over xBC slice of zxbcdt
// ---------------------------------------------------------------------------
__global__ void mamba2_conv_silu(const float* __restrict__ zx,
                                 const float* __restrict__ conv_w,
                                 const float* __restrict__ conv_b,
                                 float* __restrict__ xbc) {
    size_t idx = (size_t)blockIdx.x * blockDim.x + threadIdx.x;
    const size_t total = (size_t)NTOK * CONV_DIM;
    if (idx >= total) return;
    const int ch = (int)(idx % CONV_DIM);
    const size_t pos = idx / CONV_DIM;           // b*4096 + l
    const int l = (int)(pos & (SEQ - 1));
    float acc = conv_b[ch];
#pragma unroll
    for (int k = 0; k < 4; ++k) {
        const int ls = l + k - 3;
        if (ls >= 0)
            acc += zx[(pos + (size_t)(ls - l)) * D_IN_PROJ + D_INNER + ch] *
                   conv_w[ch * 4 + k];
    }
    xbc[pos * CONV_DIM + ch] = siluf(acc);
}

// ---------------------------------------------------------------------------
// dt = softplus(raw + bias); per-chunk cumsum of A*dt.  One thread per (b,h,c).
// ---------------------------------------------------------------------------
__global__ void mamba2_dt_cumsum(const float* __restrict__ zx,
                                 const float* __restrict__ dt_bias,
                                 const float* __restrict__ A_log,
                                 float* __restrict__ dtg,
                                 float* __restrict__ acs) {
    const int idx = blockIdx.x * blockDim.x + threadIdx.x;
    if (idx >= BATCH * NHEADS * NCHUNK) return;
    const int c = idx & 63;
    const int h = (idx >> 6) & 31;
    const int b = idx >> 11;
    const float a = -__expf(A_log[h]);
    const float bias = dt_bias[h];
    float run = 0.f;
    const size_t abase = (((size_t)b * NHEADS + h) * NCHUNK + c) * CHUNK;
    for (int l = 0; l < CHUNK; ++l) {
        const size_t pos = (size_t)b * SEQ + c * CHUNK + l;
        float raw = zx[pos * D_IN_PROJ + (D_INNER + CONV_DIM) + h] + bias;
        float dtv = (raw > 20.f) ? raw : log1pf(__expf(raw));
        dtg[pos * NHEADS + h] = dtv;
        run += a * dtv;
        acs[abase + l] = run;
    }
}

// ---------------------------------------------------------------------------
// Per (b,c,h): Y_diag = (exp(segsum) ∘ C·B^T)·X'  and  states = (X'∘decay)^T·B
// block = 128 threads (4 waves); 64x64 matrices tiled as 16x16 WMMA f32.
// B/C tiles arrive via the Tensor Data Mover when available.
// ---------------------------------------------------------------------------
__global__ void __launch_bounds__(128)
mamba2_chunk_diag_states(const float* __restrict__ xbc,
                         const float* __restrict__ dtg,
                         const float* __restrict__ acs,
                         float* __restrict__ ybuf,
                         float* __restrict__ cstates) {
    const int h = blockIdx.x & 31;
    const int c = (blockIdx.x >> 5) & 63;
    const int b = blockIdx.x >> 11;

    __shared__ float Cl[64 * 64];   // C, later reused to hold masked M
    __shared__ float Bl[64 * 64];
    __shared__ float Xl[64 * 64];   // X' = silu(x)*dt
    __shared__ float acs_s[64];
    __shared__ float dec_s[64];

    const int tid = threadIdx.x;
    const int lane = tid & 31;
    const int wave = tid >> 5;
    const int half = lane >> 4;
    const int lm   = lane & 15;
    const int m    = wave * 16 + lm;
    const size_t rowbase = (size_t)b * SEQ + (size_t)c * CHUNK;  // token of l=0

#ifdef MAMBA2_HAS_TDM
    if (wave == 0) {   // one wave drives the TDM; EXEC ignored by tensor ops
        tdm_load_tile64(xbc + rowbase * CONV_DIM + D_INNER + D_STATE,
                        lds_off(Cl), CONV_DIM);
        tdm_load_tile64(xbc + rowbase * CONV_DIM + D_INNER,
                        lds_off(Bl), CONV_DIM);
    }
    for (int i = tid; i < 4096; i += 128) {
        const int l = i >> 6, j = i & 63;
        Xl[i] = xbc[(rowbase + l) * CONV_DIM + h * HEADDIM + j] *
                dtg[(rowbase + l) * NHEADS + h];
    }
    if (wave == 0) __builtin_amdgcn_s_wait_tensorcnt(0);
#else
    for (int i = tid; i < 4096; i += 128) {
        const int l = i >> 6, j = i & 63;
        const size_t xoff = (rowbase + l) * CONV_DIM;
        Cl[i] = xbc[xoff + D_INNER + D_STATE + j];
        Bl[i] = xbc[xoff + D_INNER + j];
        Xl[i] = xbc[xoff + h * HEADDIM + j] * dtg[(rowbase + l) * NHEADS + h];
    }
#endif
    if (tid < 64)
        acs_s[tid] = acs[(((size_t)b * NHEADS + h) * NCHUNK + c) * CHUNK + tid];
    __syncthreads();
    if (tid < 64) dec_s[tid] = __expf(acs_s[63] - acs_s[tid]);
    __syncthreads();

    // ---- Phase 1: G = C * B^T (rows 16w..16w+15), mask -> M into Cl ----
    v8f acc[4];
#pragma unroll
    for (int t = 0; t < 4; ++t) acc[t] = zero8();
    for (int k0 = 0; k0 < 64; k0 += 4) {
        const int ka = k0 + 2 * half;
        v2f a = *(const v2f*)&Cl[m * 64 + ka];
#pragma unroll
        for (int t = 0; t < 4; ++t) {
            v2f bf = *(const v2f*)&Bl[(t * 16 + lm) * 64 + ka];
            acc[t] = wmma4(a, bf, acc[t]);
        }
    }
#pragma unroll
    for (int t = 0; t < 4; ++t) {
        const int j = t * 16 + lm;
#pragma unroll
        for (int r = 0; r < 8; ++r) {
            const int i = wave * 16 + r + 8 * half;
            float v = (i >= j) ? acc[t][r] * __expf(acs_s[i] - acs_s[j]) : 0.f;
            Cl[i * 64 + j] = v;          // own row stripe only
        }
    }
    __syncthreads();

    // ---- Phase 2: Y_diag = M * X' ----
    v8f acc2[4];
#pragma unroll
    for (int t = 0; t < 4; ++t) acc2[t] = zero8();
    for (int k0 = 0; k0 < 64; k0 += 4) {
        const int ka = k0 + 2 * half;
        v2f a = *(const v2f*)&Cl[m * 64 + ka];
#pragma unroll
        for (int t = 0; t < 4; ++t) {
            const int p = t * 16 + lm;
            v2f bf = {Xl[ka * 64 + p], Xl[(ka + 1) * 64 + p]};
            acc2[t] = wmma4(a, bf, acc2[t]);
        }
    }
#pragma unroll
    for (int t = 0; t < 4; ++t) {
        const int p = t * 16 + lm;
#pragma unroll
        for (int r = 0; r < 8; ++r) {
            const int i = wave * 16 + r + 8 * half;   // local l
            ybuf[(rowbase + i) * D_INNER + h * HEADDIM + p] = acc2[t][r];
        }
    }

    // ---- Phase 3: states[p][n] = sum_l X'[l][p]*decay[l]*B[l][n] ----
    v8f acc3[4];
#pragma unroll
    for (int t = 0; t < 4; ++t) acc3[t] = zero8();
    for (int k0 = 0; k0 < 64; k0 += 4) {
        const int ka = k0 + 2 * half;
        v2f a = {Xl[ka * 64 + m] * dec_s[ka], Xl[(ka + 1) * 64 + m] * dec_s[ka + 1]};
#pragma unroll
        for (int t = 0; t < 4; ++t) {
            const int n = t * 16 + lm;
            v2f bf = {Bl[ka * 64 + n], Bl[(ka + 1) * 64 + n]};
            acc3[t] = wmma4(a, bf, acc3[t]);
        }
    }
    const size_t sbase = (size_t)blockIdx.x * 4096;   // [b][c][h][p][n]
#pragma unroll
    for (int t = 0; t < 4; ++t) {
        const int n = t * 16 + lm;
#pragma unroll
        for (int r = 0; r < 8; ++r) {
            const int p = wave * 16 + r + 8 * half;
            cstates[sbase + p * 64 + n] = acc3[t][r];
        }
    }
}

// ---------------------------------------------------------------------------
// Inter-chunk recurrence: S_0 = init; state_in[c] = S; S = exp(acs_last)*S + cs[c]
// ---------------------------------------------------------------------------
__global__ void __launch_bounds__(256)
mamba2_state_recurrence(const float* __restrict__ init_states,
                        const float* __restrict__ acs,
                        const float* __restrict__ cstates,
                        float* __restrict__ sin_) {
    const int bh = blockIdx.x;
    const int b = bh >> 5, h = bh & 31;
    const int tid = threadIdx.x;
    float s[16];
#pragma unroll
    for (int i = 0; i < 16; ++i)
        s[i] = init_states[(size_t)h * 4096 + tid + i * 256];
    for (int c = 0; c < NCHUNK; ++c) {
        const float dchunk =
            __expf(acs[(((size_t)b * NHEADS + h) * NCHUNK + c) * CHUNK + 63]);
        const size_t base = (((size_t)b * NCHUNK + c) * NHEADS + h) * 4096;
#pragma unroll
        for (int i = 0; i < 16; ++i) {
            const int e = tid + i * 256;
            sin_[base + e] = s[i];
            s[i] = dchunk * s[i] + cstates[base + e];
        }
    }
}

// ---------------------------------------------------------------------------
// Per (b,c,h): Y += exp(acs[l]) * (C · state_in^T) + D[h]*x
// ---------------------------------------------------------------------------
__global__ void __launch_bounds__(128)
mamba2_chunk_off(const float* __restrict__ xbc,
                 const float* __restrict__ acs,
                 const float* __restrict__ sin_,
                 const float* __restrict__ Dvec,
                 float* __restrict__ ybuf) {
    const int h = blockIdx.x & 31;
    const int c = (blockIdx.x >> 5) & 63;
    const int b = blockIdx.x >> 11;

    __shared__ float Cl[64 * 64];
    __shared__ float Sl[64 * 64];   // state_in, [p][n]
    __shared__ float acs_s[64];

    const int tid = threadIdx.x;
    const int lane = tid & 31;
    const int wave = tid >> 5;
    const int half = lane >> 4;
    const int lm   = lane & 15;
    const int m    = wave * 16 + lm;
    const size_t rowbase = (size_t)b * SEQ + (size_t)c * CHUNK;
    const size_t sbase = (size_t)blockIdx.x * 4096;

#ifdef MAMBA2_HAS_TDM
    if (wave == 0) {
        tdm_load_tile64(xbc + rowbase * CONV_DIM + D_INNER + D_STATE,
                        lds_off(Cl), CONV_DIM);
        tdm_load_tile64(sin_ + sbase, lds_off(Sl), 64);
        __builtin_amdgcn_s_wait_tensorcnt(0);
    }
#else
    for (int i = tid; i < 4096; i += 128) {
        const int l = i >> 6, j = i & 63;
        Cl[i] = xbc[(rowbase + l) * CONV_DIM + D_INNER + D_STATE + j];
        Sl[i] = sin_[sbase + i];
    }
#endif
    if (tid < 64)
        acs_s[tid] = acs[(((size_t)b * NHEADS + h) * NCHUNK + c) * CHUNK + tid];
    __syncthreads();

    v8f acc[4];
#pragma unroll
    for (int t = 0; t < 4; ++t) acc[t] = zero8();
    for (int k0 = 0; k0 < 64; k0 += 4) {      // k runs over n (D_STATE)
        const int ka = k0 + 2 * half;
        v2f a = *(const v2f*)&Cl[m * 64 + ka];
#pragma unroll
        for (int t = 0; t < 4; ++t) {
            const int p = t * 16 + lm;
            v2f bf = *(const v2f*)&Sl[p * 64 + ka];   // B[k=n][p] = S[p][n]
            acc[t] = wmma4(a, bf, acc[t]);
        }
    }
    const float Dh = Dvec[h];
#pragma unroll
    for (int t = 0; t < 4; ++t) {
        const int p = t * 16 + lm;
#pragma unroll
        for (int r = 0; r < 8; ++r) {
            const int l = wave * 16 + r + 8 * half;
            const float sdo = __expf(acs_s[l]);
            const size_t yoff = (rowbase + l) * D_INNER + h * HEADDIM + p;
            const float xv = xbc[(rowbase + l) * CONV_DIM + h * HEADDIM + p];
            ybuf[yoff] = ybuf[yoff] + acc[t][r] * sdo + Dh * xv;
        }
    }
}

// ---------------------------------------------------------------------------
// y = Y * silu(z); RMSNorm over 2048 with norm_w.  One block (256 thr) per token.
// ---------------------------------------------------------------------------
__global__ void __launch_bounds__(256)
mamba2_gate_rmsnorm(const float* __restrict__ zx,
                    const float* __restrict__ norm_w,
                    float* __restrict__ ybuf) {
    const int row = blockIdx.x;
    const int tid = threadIdx.x;
    const float* zr = zx + (size_t)row * D_IN_PROJ;     // z = first 2048
    float* yr = ybuf + (size_t)row * D_INNER;
    float vals[8];
    float sum = 0.f;
#pragma unroll
    for (int i = 0; i < 8; ++i) {
        const int j = tid + i * 256;
        const float v = yr[j] * siluf(zr[j]);
        vals[i] = v;
        sum += v * v;
    }
    __shared__ float red[256];
    red[tid] = sum;
    __syncthreads();
    for (int s2 = 128; s2 > 0; s2 >>= 1) {
        if (tid < s2) red[tid] += red[tid + s2];
        __syncthreads();
    }
    const float scale = rsqrtf(red[0] / (float)D_INNER + 1e-5f);
#pragma unroll
    for (int i = 0; i < 8; ++i) {
        const int j = tid + i * 256;
        yr[j] = vals[i] * scale * norm_w[j];
    }
}

// ---------------------------------------------------------------------------
extern "C" void kernel_launch(void* const* d_in, const int* in_sizes, int n_in,
                              void* d_out, int out_size, void* d_ws, size_t ws_size,
                              hipStream_t stream) {
    (void)in_sizes; (void)n_in; (void)out_size; (void)ws_size;
    const float* u           = (const float*)d_in[0];
    const float* in_proj_w   = (const float*)d_in[1];
    const float* conv_w      = (const float*)d_in[2];
    const float* conv_b      = (const float*)d_in[3];
    const float* init_states = (const float*)d_in[4];
    const float* dt_bias     = (const float*)d_in[5];
    const float* A_log       = (const float*)d_in[6];
    const float* Dvec        = (const float*)d_in[7];
    const float* norm_w      = (const float*)d_in[8];
    const float* out_proj_w  = (const float*)d_in[9];
    float* out = (float*)d_out;

    float* ws = (float*)d_ws;
    size_t o = 0;
    float* zx   = ws + o; o += (size_t)NTOK * D_IN_PROJ;            // 69.7M
    float* xbc  = ws + o; o += (size_t)NTOK * CONV_DIM;             // 35.7M
    float* dtg  = ws + o; o += (size_t)NTOK * NHEADS;               // 0.5M
    float* acs  = ws + o; o += (size_t)BATCH * NHEADS * NCHUNK * CHUNK; // 0.5M
    float* cst  = ws + o; o += (size_t)BATCH * NCHUNK * NHEADS * 4096;  // 33.6M
    float* sin_ = ws + o; o += (size_t)BATCH * NCHUNK * NHEADS * 4096;  // 33.6M
    float* ybuf = ws + o; o += (size_t)NTOK * D_INNER;              // 33.6M

    // 1. in_proj: zx = u @ in_proj_w^T  (M=16384, N=4256, K=1024)
    //    266 n-tiles = 2 waves * 19 blocks * 7 tiles -> no edge guards
    mamba2_gemm_f32_wmma<7><<<dim3(NTOK / 16, 19), 64, 0, stream>>>(
        u, in_proj_w, zx, NTOK, D_IN_PROJ, D_MODEL);
    // 2. conv + silu
    {
        const size_t total = (size_t)NTOK * CONV_DIM;
        mamba2_conv_silu<<<(unsigned)((total + 255) / 256), 256, 0, stream>>>(
            zx, conv_w, conv_b, xbc);
    }
    // 3. dt softplus + per-chunk cumsum
    mamba2_dt_cumsum<<<(BATCH * NHEADS * NCHUNK + 255) / 256, 256, 0, stream>>>(
        zx, dt_bias, A_log, dtg, acs);
    // 4. intra-chunk diag output + chunk states
    mamba2_chunk_diag_states<<<BATCH * NCHUNK * NHEADS, 128, 0, stream>>>(
        xbc, dtg, acs, ybuf, cst);
    // 5. inter-chunk state recurrence
    mamba2_state_recurrence<<<BATCH * NHEADS, 256, 0, stream>>>(
        init_states, acs, cst, sin_);
    // 6. off-diagonal contribution + D skip
    mamba2_chunk_off<<<BATCH * NCHUNK * NHEADS, 128, 0, stream>>>(
        xbc, acs, sin_, Dvec, ybuf);
    // 7. gate + RMSNorm
    mamba2_gate_rmsnorm<<<NTOK, 256, 0, stream>>>(zx, norm_w, ybuf);
    // 8. out_proj: out = y @ out_proj_w^T  (M=16384, N=1024, K=2048)
    //    64 n-tiles = 2 waves * 8 blocks * 4 tiles
    mamba2_gemm_f32_wmma<4><<<dim3(NTOK / 16, 8), 64, 0, stream>>>(
        ybuf, out_proj_w, out, NTOK, D_MODEL, D_INNER);
}